// ImageDataModel_48653389529645
// MI455X (gfx1250) — compile-verified
//
#include <hip/hip_runtime.h>
#include <math.h>

// ---------------------------------------------------------------------------
// Problem constants (from reference setup_inputs)
// ---------------------------------------------------------------------------
#define NN  512      // batch N
#define KK  256      // MC samples K
#define DD  784      // data dim D
#define ZZ  2        // latent dim

typedef float v2f __attribute__((ext_vector_type(2)));
typedef float v8f __attribute__((ext_vector_type(8)));

// ---------------------------------------------------------------------------
// Layer 1: h1 = elu(z @ W1^T + b1)   (K=2 contraction, pure VALU, tiny)
// ---------------------------------------------------------------------------
__global__ void dec1_kernel(const float* __restrict__ z,   // (KK, 2)
                            const float* __restrict__ W1,  // (64, 2)
                            const float* __restrict__ b1,  // (64)
                            float* __restrict__ h1)        // (KK, 64)
{
    int idx = blockIdx.x * blockDim.x + threadIdx.x;
    if (idx >= KK * 64) return;
    int m = idx >> 6;       // latent row
    int n = idx & 63;       // output unit
    float v = fmaf(z[m * 2 + 0], W1[n * 2 + 0],
               fmaf(z[m * 2 + 1], W1[n * 2 + 1], b1[n]));
    h1[idx] = v > 0.0f ? v : expm1f(v);
}

// ---------------------------------------------------------------------------
// Generic fp32 WMMA GEMM:  C(MxN) = act( A(MxK) @ Bt(NxK)^T + bias )
// One 16x16 output tile per wave32, 4 waves per block.
// Uses V_WMMA_F32_16X16X4_F32 (fp32 in, fp32 accumulate) — matches reference
// precision while running on the CDNA5 matrix pipe.
// ACT: 0 = none, 1 = ELU, 2 = sigmoid
// ---------------------------------------------------------------------------
template <int ACT>
__global__ void gemm_wmma_f32(const float* __restrict__ A,
                              const float* __restrict__ Bt,
                              const float* __restrict__ bias,  // may be null
                              float* __restrict__ C,
                              int M, int Nc, int Kd)
{
    const int wavesPerBlock = blockDim.x >> 5;
    const int tile   = blockIdx.x * wavesPerBlock + (threadIdx.x >> 5);
    const int tilesN = Nc >> 4;
    const int totalTiles = (M >> 4) * tilesN;
    if (tile >= totalTiles) return;          // uniform per wave

    const int lane = threadIdx.x & 31;
    const int tm = (tile / tilesN) << 4;
    const int tn = (tile % tilesN) << 4;
    const int half = lane >> 4;              // lane group 0/1
    const int l16  = lane & 15;
    const int m = tm + l16;                  // A row for this lane
    const int n = tn + l16;                  // B row (= C col) for this lane

    v8f acc = {};
    const float* __restrict__ aRow = A  + (size_t)m * Kd;
    const float* __restrict__ bRow = Bt + (size_t)n * Kd;

    for (int k = 0; k < Kd; k += 4) {
        const int kk = k + (half << 1);      // lanes 0-15: K=k,k+1 ; 16-31: K=k+2,k+3
        v2f a, b;
        a.x = aRow[kk];
        a.y = aRow[kk + 1];
        b.x = bRow[kk];
        b.y = bRow[kk + 1];
        acc = __builtin_amdgcn_wmma_f32_16x16x4_f32(
            false, a, false, b, (short)0, acc, false, false);
    }

    float bv = (bias != nullptr) ? bias[n] : 0.0f;
#pragma unroll
    for (int r = 0; r < 8; ++r) {
        int mr = tm + r + (half << 3);       // C layout: M = r + 8*(lane/16)
        float v = acc[r] + bv;
        if (ACT == 1)      v = v > 0.0f ? v : expm1f(v);
        else if (ACT == 2) v = 1.0f / (1.0f + expf(-v));
        C[(size_t)mr * Nc + n] = v;
    }
}

// ---------------------------------------------------------------------------
// Row squared-norm: out[row] = sum_c X[row,c]^2
// ---------------------------------------------------------------------------
__global__ void rownorm_kernel(const float* __restrict__ X,
                               float* __restrict__ out, int cols)
{
    __shared__ float red[256];
    int row = blockIdx.x;
    float s = 0.0f;
    for (int c = threadIdx.x; c < cols; c += blockDim.x) {
        float v = X[(size_t)row * cols + c];
        s = fmaf(v, v, s);
    }
    red[threadIdx.x] = s;
    __syncthreads();
    for (int off = blockDim.x >> 1; off > 0; off >>= 1) {
        if ((int)threadIdx.x < off) red[threadIdx.x] += red[threadIdx.x + off];
        __syncthreads();
    }
    if (threadIdx.x == 0) out[row] = red[0];
}

// ---------------------------------------------------------------------------
// ll[n,k] = log_norm - 0.5 * (|X_n|^2 + |P_k|^2 - 2*S[n,k]) / (s + 0.001)
// ---------------------------------------------------------------------------
__global__ void ll_kernel(const float* __restrict__ S,
                          const float* __restrict__ normX,
                          const float* __restrict__ normP,
                          const float* __restrict__ log_std,
                          float* __restrict__ LL)
{
    int idx = blockIdx.x * blockDim.x + threadIdx.x;
    if (idx >= NN * KK) return;
    int n = idx >> 8;     // KK == 256
    int k = idx & 255;
    float ls = log_std[0];
    float sig = expf(ls);
    const float LOG_2PI = logf(2.0f * 3.14159265359f);
    float log_norm = -0.5f * (float)DD * LOG_2PI - 0.5f * ((float)DD * ls);
    float quad = (normX[n] + normP[k] - 2.0f * S[idx]) / (sig + 0.001f);
    LL[idx] = log_norm - 0.5f * quad;
}

// ---------------------------------------------------------------------------
// Global max over LL (two stages)
// ---------------------------------------------------------------------------
__global__ void max_stage1(const float* __restrict__ LL, float* __restrict__ part,
                           int total)
{
    __shared__ float red[256];
    float m = -3.402823466e38f;
    for (int i = blockIdx.x * blockDim.x + threadIdx.x; i < total;
         i += gridDim.x * blockDim.x)
        m = fmaxf(m, LL[i]);
    red[threadIdx.x] = m;
    __syncthreads();
    for (int off = blockDim.x >> 1; off > 0; off >>= 1) {
        if ((int)threadIdx.x < off)
            red[threadIdx.x] = fmaxf(red[threadIdx.x], red[threadIdx.x + off]);
        __syncthreads();
    }
    if (threadIdx.x == 0) part[blockIdx.x] = red[0];
}

__global__ void max_stage2(const float* __restrict__ part, float* __restrict__ m,
                           int count)
{
    __shared__ float red[256];
    float v = (threadIdx.x < (unsigned)count) ? part[threadIdx.x] : -3.402823466e38f;
    red[threadIdx.x] = v;
    __syncthreads();
    for (int off = blockDim.x >> 1; off > 0; off >>= 1) {
        if ((int)threadIdx.x < off)
            red[threadIdx.x] = fmaxf(red[threadIdx.x], red[threadIdx.x + off]);
        __syncthreads();
    }
    if (threadIdx.x == 0) m[0] = red[0];
}

// ---------------------------------------------------------------------------
// Per-row ell[n]:  iw cancels the prior, leaving
//   ell[n] = sum_k exp(ll_s)*ll / (K * evidence_s[n]),
//   evidence_s[n] = sum_k exp(ll_s)/K    =>  softmax_k(ll)·ll
// One 256-thread block per row (K == 256, one element per thread).
// ---------------------------------------------------------------------------
__global__ void row_ell_kernel(const float* __restrict__ LL,
                               const float* __restrict__ mptr,
                               float* __restrict__ ell)
{
    __shared__ float se[256];
    __shared__ float sev[256];
    int n = blockIdx.x;
    float m = mptr[0];
    float v = LL[(size_t)n * KK + threadIdx.x];
    float e = expf(v - m);
    se[threadIdx.x]  = e;
    sev[threadIdx.x] = e * v;
    __syncthreads();
    for (int off = 128; off > 0; off >>= 1) {
        if ((int)threadIdx.x < off) {
            se[threadIdx.x]  += se[threadIdx.x + off];
            sev[threadIdx.x] += sev[threadIdx.x + off];
        }
        __syncthreads();
    }
    if (threadIdx.x == 0) ell[n] = sev[0] / se[0];
}

__global__ void final_mean_kernel(const float* __restrict__ ell,
                                  float* __restrict__ out)
{
    __shared__ float red[512];
    red[threadIdx.x] = ell[threadIdx.x];
    __syncthreads();
    for (int off = 256; off > 0; off >>= 1) {
        if ((int)threadIdx.x < off) red[threadIdx.x] += red[threadIdx.x + off];
        __syncthreads();
    }
    if (threadIdx.x == 0) out[0] = red[0] / (float)NN;
}

// ---------------------------------------------------------------------------
// Launcher
// ---------------------------------------------------------------------------
extern "C" void kernel_launch(void* const* d_in, const int* in_sizes, int n_in,
                              void* d_out, int out_size, void* d_ws, size_t ws_size,
                              hipStream_t stream)
{
    (void)in_sizes; (void)n_in; (void)out_size; (void)ws_size;
    const float* X       = (const float*)d_in[0];   // (512, 784)
    const float* z       = (const float*)d_in[1];   // (256, 2)
    const float* log_std = (const float*)d_in[2];   // (1,)
    const float* W1 = (const float*)d_in[3];  const float* b1 = (const float*)d_in[4];
    const float* W2 = (const float*)d_in[5];  const float* b2 = (const float*)d_in[6];
    const float* W3 = (const float*)d_in[7];  const float* b3 = (const float*)d_in[8];
    const float* W4 = (const float*)d_in[9];  const float* b4 = (const float*)d_in[10];
    float* out = (float*)d_out;

    // Workspace layout (floats)
    float* ws    = (float*)d_ws;
    float* h1    = ws;                    // 256*64   = 16384
    float* h2    = h1 + KK * 64;          // 256*128  = 32768
    float* h3    = h2 + KK * 128;         // 256*256  = 65536
    float* P     = h3 + KK * 256;         // 256*784  = 200704
    float* S     = P  + KK * DD;          // 512*256  = 131072 (reused as LL)
    float* normX = S  + NN * KK;          // 512
    float* normP = normX + NN;            // 256
    float* pmax  = normP + KK;            // 256
    float* gmax  = pmax + 256;            // 1
    float* ell   = gmax + 1;              // 512

    // --- decoder ---
    dec1_kernel<<<(KK * 64 + 255) / 256, 256, 0, stream>>>(z, W1, b1, h1);

    // h2 = elu(h1 @ W2^T + b2): tiles = (256/16)*(128/16) = 128
    gemm_wmma_f32<1><<<128 / 4, 128, 0, stream>>>(h1, W2, b2, h2, KK, 128, 64);
    // h3 = elu(h2 @ W3^T + b3): tiles = 16*16 = 256
    gemm_wmma_f32<1><<<256 / 4, 128, 0, stream>>>(h2, W3, b3, h3, KK, 256, 128);
    // P = sigmoid(h3 @ W4^T + b4): tiles = 16*49 = 784
    gemm_wmma_f32<2><<<784 / 4, 128, 0, stream>>>(h3, W4, b4, P, KK, DD, 256);

    // --- cross term S = X @ P^T : tiles = 32*16 = 512 ---
    gemm_wmma_f32<0><<<512 / 4, 128, 0, stream>>>(X, P, nullptr, S, NN, KK, DD);

    // --- row norms ---
    rownorm_kernel<<<NN, 256, 0, stream>>>(X, normX, DD);
    rownorm_kernel<<<KK, 256, 0, stream>>>(P, normP, DD);

    // --- log-likelihood matrix (in place over S) ---
    ll_kernel<<<(NN * KK + 255) / 256, 256, 0, stream>>>(S, normX, normP, log_std, S);

    // --- global max ---
    max_stage1<<<256, 256, 0, stream>>>(S, pmax, NN * KK);
    max_stage2<<<1, 256, 0, stream>>>(pmax, gmax, 256);

    // --- softmax-weighted ll per row, then mean ---
    row_ell_kernel<<<NN, 256, 0, stream>>>(S, gmax, ell);
    final_mean_kernel<<<1, 512, 0, stream>>>(ell, out);
}